// SplineConv_6554120094020
// MI455X (gfx1250) — compile-verified
//
#include <hip/hip_runtime.h>

// ---------------------------------------------------------------------------
// SplineConv as GEMM on CDNA5 (gfx1250):
//   Out(32768 x 512) = W(32768 x 28) @ B(28 x 512),  B = Ctrl reshaped/transposed
// W row i holds the 9 nonzero bi-quadratic B-spline weights (K padded 25->28).
// Memory-bound by the 67MB output store (~3us floor @ 23.3 TB/s); compute via
// V_WMMA_F32_16X16X4_F32 (exact f32, 7 K-steps per 16x16 tile).
// ---------------------------------------------------------------------------

typedef float v2f __attribute__((ext_vector_type(2)));
typedef float v8f __attribute__((ext_vector_type(8)));

#define NPTS 32768
#define DIM  512
#define MBLK 128
#define KPAD 28
#define WROW 28            // Wlds row stride in floats (gcd(28,64)=4 -> ok banks)
#define PSTR 1056          // Blds pair-row stride in floats: 2*512 + 32 (32-bank shift
                           // between adjacent K-pairs -> lane halves hit disjoint banks)

// De Boor (order 2) run on identity coefficients: returns the 3 basis weights
// for control indices k-2..k.  Mirrors the reference recursion exactly.
__device__ __forceinline__ void deboor_w3(const float* __restrict__ T, int k,
                                          float x, float w[3]) {
    float D[3][3] = {{1.f,0.f,0.f},{0.f,1.f,0.f},{0.f,0.f,1.f}};
#pragma unroll
    for (int r = 1; r <= 2; ++r) {
#pragma unroll
        for (int j = 2; j >= 1; --j) {
            if (j < r) break;
            int ni = k - 2 + j;
            float tn  = T[ni];
            float den = T[k + 1 - r + j] - tn;
            float a   = (x - tn) / den;
#pragma unroll
            for (int q = 0; q < 3; ++q)
                D[j][q] = (1.0f - a) * D[j - 1][q] + a * D[j][q];
        }
    }
    w[0] = D[2][0]; w[1] = D[2][1]; w[2] = D[2][2];
}

__global__ __launch_bounds__(256)
void spline_wmma_kernel(const float* __restrict__ cd,
                        const float* __restrict__ Tx,
                        const float* __restrict__ Ty,
                        const float* __restrict__ C,
                        float* __restrict__ out)
{
    __shared__ __align__(16) float Wlds[MBLK * WROW];   // 14336 B
    __shared__ __align__(16) float Blds[14 * PSTR];     // 59136 B  (total ~73 KB)

    const int tid = threadIdx.x;
    const int blk = blockIdx.x;

    // ---- stage B into LDS, pair-interleaved over K:
    //      element (r, d) -> Blds[(r>>1)*PSTR + 2*d + (r&1)]
    //      B[r][d] = Ctrl(d, r/5, r%5) = Cflat[d*25 + r]; rows 25..27 are zero pad.
    for (int idx = tid; idx < KPAD * DIM; idx += 256) {
        int d = idx / KPAD;
        int r = idx - d * KPAD;
        float v = (r < 25) ? C[d * 25 + r] : 0.0f;
        Blds[(r >> 1) * PSTR + 2 * d + (r & 1)] = v;
    }

    // ---- per-point spline weights for this block's 128 points ----
    if (tid < MBLK) {
        const int pt = blk * MBLK + tid;
        const float X = cd[2 * pt + 0];
        const float Y = cd[2 * pt + 1];

        // searchsorted(T, x, side='right') - 1, clipped to [2, 4]
        int kx = -1, ky = -1;
#pragma unroll
        for (int j = 0; j < 8; ++j) {
            kx += (Tx[j] <= X) ? 1 : 0;
            ky += (Ty[j] <= Y) ? 1 : 0;
        }
        kx = min(max(kx, 2), 4);
        ky = min(max(ky, 2), 4);

        float wx[3], wy[3];
        deboor_w3(Tx, kx, X, wx);
        deboor_w3(Ty, ky, Y, wy);

        float* wrow = &Wlds[tid * WROW];
#pragma unroll
        for (int p = 0; p < KPAD; ++p) wrow[p] = 0.0f;
#pragma unroll
        for (int a = 0; a < 3; ++a)
#pragma unroll
            for (int b = 0; b < 3; ++b)
                wrow[(kx - 2 + a) * 5 + (ky - 2 + b)] = wx[a] * wy[b];
    }
    __syncthreads();

    // ---- WMMA GEMM: wave w owns N-tiles {4w..4w+3}; 8 M-tiles per block ----
    const int wv = tid >> 5;        // wave id 0..7
    const int ln = tid & 31;        // lane
    const int lh = ln >> 4;         // lane half (0: lanes 0-15, 1: lanes 16-31)
    const int lc = ln & 15;

    // B fragments cached in registers: lane half lh carries K = 4k+2*lh, 4k+2*lh+1
    v2f bfrag[4][7];
#pragma unroll
    for (int t = 0; t < 4; ++t) {
        const int nb = wv * 64 + t * 16 + lc;
#pragma unroll
        for (int k = 0; k < 7; ++k) {
            const int kpair = 2 * k + lh;
            bfrag[t][k] = *(const v2f*)&Blds[kpair * PSTR + 2 * nb];
        }
    }

#pragma unroll
    for (int m = 0; m < 8; ++m) {
        // A fragments: lanes 0-15 hold M=0..15 K={4k,4k+1}; lanes 16-31 K={4k+2,4k+3}
        v2f afrag[7];
        const float* wbase = &Wlds[(m * 16 + lc) * WROW];
#pragma unroll
        for (int k = 0; k < 7; ++k)
            afrag[k] = *(const v2f*)&wbase[4 * k + 2 * lh];

#pragma unroll
        for (int t = 0; t < 4; ++t) {
            v8f acc = {};
#pragma unroll
            for (int k = 0; k < 7; ++k)
                acc = __builtin_amdgcn_wmma_f32_16x16x4_f32(
                    /*neg_a=*/false, afrag[k],
                    /*neg_b=*/false, bfrag[t][k],
                    /*c_mod=*/(short)0, acc,
                    /*reuse_a=*/false, /*reuse_b=*/false);

            // D layout: VGPR j -> row M = j + 8*lh, col N = lc
            const int ch  = wv * 64 + t * 16 + lc;
            const int row = blk * MBLK + m * 16 + lh * 8;
            float* o = out + row * DIM + ch;
#pragma unroll
            for (int j = 0; j < 8; ++j)
                o[j * DIM] = acc[j];
        }
    }
}

extern "C" void kernel_launch(void* const* d_in, const int* in_sizes, int n_in,
                              void* d_out, int out_size, void* d_ws, size_t ws_size,
                              hipStream_t stream) {
    const float* cd = (const float*)d_in[0];   // (32768, 2)
    const float* Tx = (const float*)d_in[1];   // (8,)
    const float* Ty = (const float*)d_in[2];   // (8,)
    const float* C  = (const float*)d_in[3];   // (32,16,5,5) == (512,25) flat
    float* out = (float*)d_out;                // (32768, 512) flat

    dim3 grid(NPTS / MBLK);   // 256 blocks
    dim3 block(256);          // 8 wave32
    spline_wmma_kernel<<<grid, block, 0, stream>>>(cd, Tx, Ty, C, out);
}